// QuantumCircuitSimulator2_79207786873077
// MI455X (gfx1250) — compile-verified
//
#include <hip/hip_runtime.h>

// ---------------------------------------------------------------------------
// Quantum circuit simulator, 24 qubits, 8 steps of 4-qubit (16x16) complex
// gates. State (2 x 2^24 fp32 = 128 MB) is kept IN PLACE in a single buffer
// so it stays resident in the 192 MB L2 across all 8 steps; each wave32 owns
// 16 independent groups and does the 16x16 complex matmul with
// V_WMMA_F32_16X16X4_F32 chained over K (16 WMMAs per tile).
// ---------------------------------------------------------------------------

typedef __attribute__((ext_vector_type(2))) float v2f;
typedef __attribute__((ext_vector_type(8))) float v8f;

#define NQ 24
#define NSTATE (1u << NQ)           // 2^24 amplitudes
#define NGROUPS (1u << (NQ - 4))    // 2^20 groups of 16
#define NTILES (NGROUPS / 16u)      // 65536 wave-tiles

#define WMMA4(A, B, C) \
  __builtin_amdgcn_wmma_f32_16x16x4_f32(false, (A), false, (B), (short)0, (C), false, false)

__global__ __launch_bounds__(256) void qsim_step_kernel(
    float* __restrict__ state,          // [2][2^24], in-place update
    const int* __restrict__ targets,    // [8][4]
    const float* __restrict__ gates,    // [8][2][16][16]
    int step)
{
  const unsigned lane = threadIdx.x & 31u;
  const unsigned waveInBlock = threadIdx.x >> 5;
  const unsigned tile = blockIdx.x * 8u + waveInBlock;   // 0 .. 65535

  // ---- per-step target qubits (wave-uniform) ----
  const int t0 = targets[step * 4 + 0];
  const int t1 = targets[step * 4 + 1];
  const int t2 = targets[step * 4 + 2];
  const int t3 = targets[step * 4 + 3];

  // sorted copy (ascending) for the zero-insertion expansion
  int s0 = t0, s1 = t1, s2 = t2, s3 = t3, tv;
  if (s0 > s1) { tv = s0; s0 = s1; s1 = tv; }
  if (s2 > s3) { tv = s2; s2 = s3; s3 = tv; }
  if (s0 > s2) { tv = s0; s0 = s2; s2 = tv; }
  if (s1 > s3) { tv = s1; s1 = s3; s3 = tv; }
  if (s1 > s2) { tv = s1; s1 = s2; s2 = tv; }

  float* __restrict__ re = state;
  float* __restrict__ im = state + NSTATE;
  const float* __restrict__ gr = gates + (size_t)step * 2 * 256;
  const float* __restrict__ gi = gr + 256;

  const unsigned col = lane & 15u;      // N for B/C/D, M row for A bases
  const unsigned hk  = (lane >> 4) << 1; // K sub-offset: 0 or 2

  // spread 4-bit amplitude index into target bit positions (original order:
  // bit j of k -> qubit targets[step][j])
  auto spread = [&](unsigned k) -> unsigned {
    return ((k & 1u) << t0) | (((k >> 1) & 1u) << t1) |
           (((k >> 2) & 1u) << t2) | (((k >> 3) & 1u) << t3);
  };

  // ---- group base address for row M = col of this tile ----
  unsigned g = tile * 16u + col;
  // insert zero bits at sorted target positions s0<s1<s2<s3
  g = ((g >> s0) << (s0 + 1)) | (g & ((1u << s0) - 1u));
  g = ((g >> s1) << (s1 + 1)) | (g & ((1u << s1) - 1u));
  g = ((g >> s2) << (s2 + 1)) | (g & ((1u << s2) - 1u));
  g = ((g >> s3) << (s3 + 1)) | (g & ((1u << s3) - 1u));
  const unsigned base = g;

  // ---- B operands: Bgr[k][n] = gr[n][k] (gate transposed), + gi and -gi ----
  v2f Bgr[4], Bgi[4], Bgin[4];
#pragma unroll
  for (int kk = 0; kk < 4; ++kk) {
    const unsigned K = 4u * kk + hk;
    Bgr[kk][0]  = gr[col * 16u + K];
    Bgr[kk][1]  = gr[col * 16u + K + 1u];
    Bgi[kk][0]  = gi[col * 16u + K];
    Bgi[kk][1]  = gi[col * 16u + K + 1u];
    Bgin[kk][0] = -Bgi[kk][0];
    Bgin[kk][1] = -Bgi[kk][1];
  }

  // ---- A operands: gather 16 amplitudes of this lane's row (re and im) ----
  v2f Are[4], Aim[4];
#pragma unroll
  for (int kk = 0; kk < 4; ++kk) {
    const unsigned K = 4u * kk + hk;
    const unsigned o0 = base + spread(K);
    const unsigned o1 = base + spread(K + 1u);
    Are[kk][0] = re[o0];
    Are[kk][1] = re[o1];
    Aim[kk][0] = im[o0];
    Aim[kk][1] = im[o1];
  }

  // ---- complex 16x16 matmul via 16 fp32 WMMAs ----
  // nr = Sr*Gr^T - Si*Gi^T ; ni = Sr*Gi^T + Si*Gr^T
  v8f cr = {};
  v8f ci = {};
#pragma unroll
  for (int kk = 0; kk < 4; ++kk) cr = WMMA4(Are[kk], Bgr[kk], cr);
#pragma unroll
  for (int kk = 0; kk < 4; ++kk) cr = WMMA4(Aim[kk], Bgin[kk], cr);
#pragma unroll
  for (int kk = 0; kk < 4; ++kk) ci = WMMA4(Are[kk], Bgi[kk], ci);
#pragma unroll
  for (int kk = 0; kk < 4; ++kk) ci = WMMA4(Aim[kk], Bgr[kk], ci);

  // ---- scatter stores: lane holds column N=col for rows M=r+8*(lane>>4) ----
  const unsigned offS = spread(col);
  const unsigned mhi = (lane >> 4) * 8u;
#pragma unroll
  for (int r = 0; r < 8; ++r) {
    const int m = (int)(r + mhi);
    // fetch row m's group base from lane m (lanes 0..15 hold rows 0..15)
    const unsigned bm =
        (unsigned)__builtin_amdgcn_ds_bpermute(m << 2, (int)base);
    re[bm + offS] = cr[r];
    im[bm + offS] = ci[r];
  }
}

extern "C" void kernel_launch(void* const* d_in, const int* in_sizes, int n_in,
                              void* d_out, int out_size, void* d_ws, size_t ws_size,
                              hipStream_t stream) {
  (void)in_sizes; (void)n_in; (void)d_ws; (void)ws_size;
  const int* targets = (const int*)d_in[1];
  const float* gates = (const float*)d_in[2];
  float* out = (float*)d_out;

  // copy input state into d_out; all 8 steps update it in place (the 128 MB
  // working set stays L2-resident across steps).
  hipMemcpyAsync(d_out, d_in[0], (size_t)out_size * sizeof(float),
                 hipMemcpyDeviceToDevice, stream);

  for (int step = 0; step < 8; ++step) {
    qsim_step_kernel<<<NTILES / 8, 256, 0, stream>>>(out, targets, gates, step);
  }
}